// KResampleRenderer_31069793419365
// MI455X (gfx1250) — compile-verified
//
#include <hip/hip_runtime.h>
#include <hip/hip_bf16.h>

// KResampleRenderer for MI455X (gfx1250), wave32.
//
// out[c, oy, ox] (complex64, interleaved) = 3.0 * sum wy*wx * kim[c, yidx, xidx],
// kim = fftshift(input), output ifftshifted. Separable Catmull-Rom, u step 0.2221
// => a 16-wide output span touches <= 8 input samples per axis.
//
// One wave computes a 16x16 output tile for 4 consecutive channels via chained
// V_WMMA_F32_16X16X4_F32 GEMMs per channel (8 WMMAs: x-pass then y-pass, re+im).
// Tricks:
//  * Dense weights computed directly as cubicw(distance) (cubic kernel is 0 for
//    |d|>=2 and all taps are in-bounds) -> no floor / tap-select per lane.
//  * Stage-A A-matrix rows 8..15 duplicate rows 0..7, so stage-A's D registers
//    hold H[g][n] identically in BOTH wave halves -> stage-B B-operand is a pure
//    register select (no ds_bpermute / LDS traffic).
//  * RES_RATIO (3.0) folded into the Wy weights -> no per-store scaling.
//  * ifftshift (+-768) is 16-aligned in BOTH axes -> a tile's 8 store addresses
//    are base + g*12288B (24-bit immediate offsets); one u64 base per channel.
//  * Weights + all addressing hoisted out of the 4-channel loop.
// Roofline: 302 MB of stores @ 23.3 TB/s ~ 13 us; ~11M total instructions keeps
// issue well below that. Stores are 8B/lane, 128B-contiguous per 16-lane row.

typedef float v2f __attribute__((ext_vector_type(2)));
typedef float v8f __attribute__((ext_vector_type(8)));

#define WMMA4(A, B, C) \
  __builtin_amdgcn_wmma_f32_16x16x4_f32(false, (A), false, (B), (short)0, (C), false, false)

__device__ __forceinline__ float cubicw(float x) {
  float ax = fabsf(x);
  float f1 = 1.0f + ax * ax * (1.5f * ax - 2.5f);
  float b  = 2.0f - ax;
  float f2 = -0.5f * (ax - 1.0f) * b * b;
  return ax <= 1.0f ? f1 : (ax < 2.0f ? f2 : 0.0f);
}

// u(g): output grid position g in input-index space. k_in0=-10, 1/h=51.15.
// Range is exactly [341, 682] -> all 4 taps always in [0,1024).
__device__ __forceinline__ double ucoord(int g) {
  return ((-0.5 + (double)g * (1.0 / 1535.0)) * (1.0 / 0.15) + 10.0) * 51.15;
}

__global__ __launch_bounds__(256) void kresample_wmma(const float* __restrict__ re,
                                                      const float* __restrict__ im,
                                                      float* __restrict__ out) {
  const int lane = threadIdx.x & 31;
  const int n    = lane & 15;          // matrix column / row index held by this lane
  const int kb   = (lane >> 4) << 1;   // per-half K base: 0 (lanes 0-15) or 2 (16-31)
  const bool lo  = lane < 16;

  // 96*96 tiles * 4 channel-groups; 8 waves per block, exact fit (4608 blocks).
  int wid = blockIdx.x * 8 + (threadIdx.x >> 5);
  int tX  = wid % 96;
  int t2  = wid / 96;
  int tY  = t2 % 96;
  int c0  = (t2 / 96) * 4;             // first of 4 channels for this wave
  int gx0 = tX * 16, gy0 = tY * 16;

  // Tile footprint bases: 8 input columns/rows cover all taps of 16 outputs.
  int colBase = (int)floor(ucoord(gx0)) - 1;
  int rowBase = (int)floor(ucoord(gy0)) - 1;

  // ---- weights as direct cubic(distance); B = Wx^T, A(stage B) = 3*Wy ----
  double ux = ucoord(gx0 + n);
  float fx = (float)(ux - (double)(colBase + kb));
  v2f b0  = { cubicw(fx),        cubicw(fx - 1.0f) };
  v2f b1  = { cubicw(fx - 4.0f), cubicw(fx - 5.0f) };

  double uy = ucoord(gy0 + n);
  float fy = (float)(uy - (double)(rowBase + kb));
  v2f aw0 = { 3.0f * cubicw(fy),        3.0f * cubicw(fy - 1.0f) };  // RES_RATIO folded in
  v2f aw1 = { 3.0f * cubicw(fy - 4.0f), 3.0f * cubicw(fy - 5.0f) };

  // ---- gather addressing (fftshift: (idx+512)&1023); rows 8..15 duplicate 0..7 ----
  int ym = (rowBase + (n & 7) + 512) & 1023;
  size_t rowIn = (size_t)ym * 1024u;
  int x0 = (colBase + kb + 0 + 512) & 1023;
  int x1 = (colBase + kb + 1 + 512) & 1023;
  int x2 = (colBase + kb + 4 + 512) & 1023;
  int x3 = (colBase + kb + 5 + 512) & 1023;
  const float* pr = re + (size_t)c0 * 1048576u + rowIn;
  const float* pi = im + (size_t)c0 * 1048576u + rowIn;

  // ---- store base: wrap branch is uniform per tile in BOTH axes (768 = 48*16),
  //      so the 8 row addresses are base + g*(1536 complex) with immediate offsets.
  int mx     = gx0 + n + ((gx0 >= 768) ? -768 : 768);
  int myBase = gy0 + (lo ? 0 : 8) + ((gy0 >= 768) ? -768 : 768);
  const size_t OUTPLANE = 2u * 1536u * 1536u;    // floats per channel plane
  float* pbase = out + (size_t)c0 * OUTPLANE + 2u * ((size_t)myBase * 1536u + (unsigned)mx);

#pragma unroll
  for (int cc = 0; cc < 4; ++cc) {
    // A = G (8x8 footprint, rows duplicated into M=8..15)
    v2f a0r = { pr[x0], pr[x1] };
    v2f a1r = { pr[x2], pr[x3] };
    v2f a0i = { pi[x0], pi[x1] };
    v2f a1i = { pi[x2], pi[x3] };

    // Stage A: H = G x Wx^T (K=8 -> 2 k-steps), real & imag.
    v8f hr = {0.f,0.f,0.f,0.f,0.f,0.f,0.f,0.f};
    v8f hi = {0.f,0.f,0.f,0.f,0.f,0.f,0.f,0.f};
    hr = WMMA4(a0r, b0, hr);
    hr = WMMA4(a1r, b1, hr);
    hi = WMMA4(a0i, b0, hi);
    hi = WMMA4(a1i, b1, hi);

    // Stage B B-operand straight from own registers: hr[g] == H[g][n] in both halves.
    v2f bh0r = { lo ? hr[0] : hr[2], lo ? hr[1] : hr[3] };
    v2f bh1r = { lo ? hr[4] : hr[6], lo ? hr[5] : hr[7] };
    v2f bh0i = { lo ? hi[0] : hi[2], lo ? hi[1] : hi[3] };
    v2f bh1i = { lo ? hi[4] : hi[6], lo ? hi[5] : hi[7] };

    // Stage B: OUT = (3*Wy) x H (K=8 -> 2 k-steps), real & imag.
    v8f qr = {0.f,0.f,0.f,0.f,0.f,0.f,0.f,0.f};
    v8f qi = {0.f,0.f,0.f,0.f,0.f,0.f,0.f,0.f};
    qr = WMMA4(aw0, bh0r, qr);
    qr = WMMA4(aw1, bh1r, qr);
    qi = WMMA4(aw0, bh0i, qi);
    qi = WMMA4(aw1, bh1i, qi);

    // 8 stores at base + g*12288 bytes (folds into 24-bit immediate offsets).
#pragma unroll
    for (int g = 0; g < 8; ++g) {
      float2 v;
      v.x = qr[g];
      v.y = qi[g];
      *(float2*)(pbase + (size_t)(g * 3072)) = v;
    }

    pr += 1048576u;                     // next channel plane (1024*1024 floats)
    pi += 1048576u;
    pbase += OUTPLANE;
  }
}

extern "C" void kernel_launch(void* const* d_in, const int* in_sizes, int n_in,
                              void* d_out, int out_size, void* d_ws, size_t ws_size,
                              hipStream_t stream) {
  (void)in_sizes; (void)n_in; (void)d_ws; (void)ws_size; (void)out_size;
  const float* re = (const float*)d_in[0];
  const float* im = (const float*)d_in[1];
  float* out = (float*)d_out;           // complex64 -> interleaved float pairs
  // 96*96 tiles * 4 channel-groups / 8 waves per block = 4608 blocks
  kresample_wmma<<<dim3(4608), dim3(256), 0, stream>>>(re, im, out);
}